// YoloLayer_8108898254914
// MI455X (gfx1250) — compile-verified
//
#include <hip/hip_runtime.h>

// ---------------------------------------------------------------------------
// YOLO head post-process for MI455X (gfx1250).
// Memory-bound (~377 MB moved, ~16 us at 23.3 TB/s). Strategy:
//   * lanes along innermost WH dim -> all 85 channel reads fully coalesced
//   * conf (94% of output bytes) transposed through LDS (float4 ds_store_b128),
//     flushed with gfx1250 async store-from-LDS (ASYNCcnt) as coalesced
//     16B chunks; drained with s_wait_asynccnt
//   * sigmoid/exp via v_exp_f32 / v_rcp_f32
// No WMMA: this workload contains no matrix multiply.
// ---------------------------------------------------------------------------

#define YOLO_LOG2E 1.4426950408889634f

__device__ __forceinline__ float fast_exp(float x) {
    return __builtin_amdgcn_exp2f(x * YOLO_LOG2E);
}
__device__ __forceinline__ float fast_sigmoid(float x) {
    return __builtin_amdgcn_rcpf(1.0f + __builtin_amdgcn_exp2f(-x * YOLO_LOG2E));
}

#if defined(__AMDGCN__) && __has_builtin(__builtin_amdgcn_global_store_async_from_lds_b128)
#define HAVE_ASYNC_LDS 1
#else
#define HAVE_ASYNC_LDS 0
#endif

typedef int v4i_t __attribute__((ext_vector_type(4)));
#if HAVE_ASYNC_LDS
typedef __attribute__((address_space(1))) v4i_t* gas_v4i_t;
typedef __attribute__((address_space(3))) v4i_t* las_v4i_t;
#endif

// Flush one 16B chunk from LDS to global. Async (CDNA5 ASYNCcnt path) when
// the builtin exists; plain ds_load+global_store fallback otherwise.
__device__ __forceinline__ void stage_store_b128(float* gdst, const float* lsrc) {
#if HAVE_ASYNC_LDS
    v4i_t* g_generic = (v4i_t*)gdst;                  // reinterpret in generic AS
    v4i_t* l_generic = (v4i_t*)lsrc;
    __builtin_amdgcn_global_store_async_from_lds_b128(
        (gas_v4i_t)g_generic,                         // addrspacecast -> global
        (las_v4i_t)l_generic,                         // addrspacecast -> LDS
        /*imm offset*/ 0, /*cpol*/ 0);
#else
    *(float4*)gdst = *(const float4*)lsrc;
#endif
}

namespace {
constexpr int kB        = 32;
constexpr int kW        = 76;
constexpr int kH        = 76;
constexpr int kWH       = kW * kH;          // 5776
constexpr int kA        = 3;                // anchors
constexpr int kNC       = 80;               // classes
constexpr int kCh       = 5 + kNC;          // 85 channels per anchor
constexpr int kTile     = 64;               // wh cells per block
constexpr int kThreads  = kTile * kA;       // 192 threads = 6 waves (wave32)
constexpr int kBlocksPB = (kWH + kTile - 1) / kTile;  // 91
constexpr int kCellsPB  = kWH * kA;         // 17328 cells per batch image
constexpr int kConfPad  = 84;               // 80 + pad: 16B-aligned rows, 4-way bank spread
} // namespace

__global__ __launch_bounds__(kThreads)
void yolo_head_kernel(const float* __restrict__ in,
                      const float* __restrict__ anchors,
                      float* __restrict__ out)
{
    __shared__ float conf_lds[kThreads * kConfPad];   // ~63 KB -> ~5 blocks/WGP

    const int b    = blockIdx.x / kBlocksPB;
    const int tile = blockIdx.x % kBlocksPB;
    const int wh0  = tile * kTile;
    const int rem  = kWH - wh0;
    const int nwh  = (rem < kTile) ? rem : kTile;     // last tile per image = 16
    const int ncell = nwh * kA;

    const int t = threadIdx.x;
    const int a = t / kTile;                          // anchor 0..2 (uniform per 2 waves)
    const int i = t % kTile;                          // wh offset -> coalesced lanes

    float* boxes_out  = out;                                          // (B, cells, 4)
    float* conf_out   = out + (size_t)kB * kCellsPB * 4;              // (B, cells, 80)
    float* scores_out = conf_out + (size_t)kB * kCellsPB * kNC;       // (B, cells)
    const size_t cells0 = (size_t)b * kCellsPB + (size_t)wh0 * kA;

    if (i < nwh) {
        const int wh = wh0 + i;
        // input: (B, a*85+c, wh); lane-consecutive wh => every load 128B/wave
        const float* p = in + ((size_t)(b * kA + a) * kCh) * kWH + wh;
        const int cell = i * kA + a;                  // output cell index in tile

        const float sobj = fast_sigmoid(p[(size_t)4 * kWH]);
        float* crow = conf_lds + cell * kConfPad;     // 336B stride, 16B aligned
        float smax = 0.0f;                            // conf values are strictly > 0
        #pragma unroll 5
        for (int c4 = 0; c4 < kNC; c4 += 4) {
            const float v0 = fast_sigmoid(p[(size_t)(5 + c4 + 0) * kWH]) * sobj;
            const float v1 = fast_sigmoid(p[(size_t)(5 + c4 + 1) * kWH]) * sobj;
            const float v2 = fast_sigmoid(p[(size_t)(5 + c4 + 2) * kWH]) * sobj;
            const float v3 = fast_sigmoid(p[(size_t)(5 + c4 + 3) * kWH]) * sobj;
            float4 vv; vv.x = v0; vv.y = v1; vv.z = v2; vv.w = v3;
            *(float4*)(crow + c4) = vv;               // one ds_store_b128 per 4 classes
            smax = fmaxf(smax, fmaxf(fmaxf(v0, v1), fmaxf(v2, v3)));
        }

        // boxes: b_xy = (sigmoid(t_xy) + grid)/[W,H]; b_wh = exp(t_wh)*anchors
        const float gx = (float)(wh / kH);            // gi (w index) pairs channel 0
        const float gy = (float)(wh % kH);            // gj (h index) pairs channel 1
        const float bx = (fast_sigmoid(p[0])            + gx) * (1.0f / (float)kW);
        const float by = (fast_sigmoid(p[(size_t)kWH])  + gy) * (1.0f / (float)kH);
        const float bw = fast_exp(p[(size_t)2 * kWH]) * anchors[a];        // (2,3): row 0 = widths
        const float bh = fast_exp(p[(size_t)3 * kWH]) * anchors[kA + a];   // row 1 = heights
        const float hw = 0.5f * bw, hh = 0.5f * bh;

        float4 box;                                   // [x1, y1, x2, y2]
        box.x = bx - hw; box.y = by - hh; box.z = bx + hw; box.w = by + hh;
        *(float4*)(boxes_out + (cells0 + (size_t)cell) * 4) = box;   // 16B store, 6% of bytes
        scores_out[cells0 + (size_t)cell] = smax;
    }

    __syncthreads();   // conf tile fully staged in LDS (waits DScnt)

    // Flush conf tile: consecutive lanes -> consecutive 16B chunks of the
    // contiguous [ncell x 80] global region => fully coalesced async stores.
    const int chunksPerCell = kNC / 4;                // 20
    const int nchunk = ncell * chunksPerCell;
    for (int j = t; j < nchunk; j += kThreads) {
        const int cell = j / chunksPerCell;
        const int c    = (j - cell * chunksPerCell) * 4;
        stage_store_b128(conf_out + (cells0 + (size_t)cell) * kNC + c,
                         conf_lds + cell * kConfPad + c);
    }

#if HAVE_ASYNC_LDS && __has_builtin(__builtin_amdgcn_s_wait_asynccnt)
    __builtin_amdgcn_s_wait_asynccnt(0);              // drain before LDS is reused
#endif
}

extern "C" void kernel_launch(void* const* d_in, const int* in_sizes, int n_in,
                              void* d_out, int out_size, void* d_ws, size_t ws_size,
                              hipStream_t stream) {
    (void)in_sizes; (void)n_in; (void)out_size; (void)d_ws; (void)ws_size;
    const float* in      = (const float*)d_in[0];
    const float* anchors = (const float*)d_in[1];
    float* out           = (float*)d_out;

    const dim3 grid(kB * kBlocksPB);   // 32 * 91 = 2912 blocks
    const dim3 block(kThreads);        // 192 threads = 6 wave32
    yolo_head_kernel<<<grid, block, 0, stream>>>(in, anchors, out);
}